// Features2im_89172110999802
// MI455X (gfx1250) — compile-verified
//
#include <hip/hip_runtime.h>
#include <cstdint>

// Problem geometry (fixed by the reference): B*C = 256 planes of 512x512 in,
// 513x513 out.
#define NPLANES 256
#define H 512
#define W 512
#define OW 513
#define BLOCK 256
#define NBANDS 16
#define BAND 33   // 16*33 = 528 >= 513

// 16-byte int vector, matching the builtin's parameter type from the
// round-1 diagnostic: 'int __vector_size__(4*sizeof(int)) __device__ *'
typedef int v4i __attribute__((vector_size(16)));
typedef __attribute__((address_space(1))) v4i g_v4i;   // global (AS1)
typedef __attribute__((address_space(3))) v4i l_v4i;   // LDS (AS3)

__device__ __forceinline__ void async_copy_b128(const float* gsrc, float* lds_dst) {
#if __has_builtin(__builtin_amdgcn_global_load_async_to_lds_b128)
  __builtin_amdgcn_global_load_async_to_lds_b128(
      (g_v4i*)(uintptr_t)gsrc,
      (l_v4i*)(uint32_t)(uintptr_t)lds_dst,
      0 /*offset*/, 0 /*cpol*/);
#else
  uint32_t l = (uint32_t)(uintptr_t)lds_dst;   // low 32 bits of flat LDS addr = LDS offset
  asm volatile("global_load_async_to_lds_b128 %0, %1, off"
               :: "v"(l), "v"(gsrc) : "memory");
#endif
}

__device__ __forceinline__ void wait_async_le2() {
#if __has_builtin(__builtin_amdgcn_s_wait_asynccnt)
  __builtin_amdgcn_s_wait_asynccnt(2);
#else
  asm volatile("s_wait_asynccnt 2" ::: "memory");
#endif
}

__device__ __forceinline__ void wait_async_le0() {
#if __has_builtin(__builtin_amdgcn_s_wait_asynccnt)
  __builtin_amdgcn_s_wait_asynccnt(0);
#else
  asm volatile("s_wait_asynccnt 0" ::: "memory");
#endif
}

// Stage one input row y (512 floats x 4 inputs = 8KB) into sb[4][W] with async
// b128 copies. 256 threads x 2 copies = 512 x 16B. Every wave issues exactly 2
// async ops (branch is wave-uniform: waves 0-3 vs 4-7), so ASYNCcnt += 2/wave.
__device__ __forceinline__ void stage_row(const float* __restrict__ in0,
                                          const float* __restrict__ in1,
                                          const float* __restrict__ in2,
                                          const float* __restrict__ in3,
                                          size_t in_off, int y,
                                          float (*sb)[W], int tid) {
  const int xi = (tid & 127) << 2;                       // 0..508, step 4
  const size_t goff = in_off + (size_t)y * W + (size_t)xi;
  if (tid < 128) {
    async_copy_b128(in0 + goff, &sb[0][xi]);
    async_copy_b128(in2 + goff, &sb[2][xi]);
  } else {
    async_copy_b128(in1 + goff, &sb[1][xi]);
    async_copy_b128(in3 + goff, &sb[3][xi]);
  }
}

__global__ __launch_bounds__(BLOCK) void Features2im_kernel(
    const float* __restrict__ in0, const float* __restrict__ in1,
    const float* __restrict__ in2, const float* __restrict__ in3,
    float* __restrict__ out) {
  __shared__ float sbuf[2][4][W];   // double-buffered staged input row
  __shared__ float tprev[OW];       // t(y-1, x) = g2(y-1,x)+g3(y-1,x-1)

  const int tid  = threadIdx.x;
  const int p    = blockIdx.x & (NPLANES - 1);
  const int band = blockIdx.x >> 8;
  const int y0   = band * BAND;
  const int y1   = (y0 + BAND < OW) ? (y0 + BAND) : OW;

  const size_t in_off  = (size_t)p * (H * W);
  const size_t out_off = (size_t)p * (OW * OW);

  int cur = 1;

  if (y0 > 0) {
    // Halo row y0-1 -> buf0, first body row y0 -> buf1 (ASYNCcnt = 4/wave).
    stage_row(in0, in1, in2, in3, in_off, y0 - 1, sbuf[0], tid);
    stage_row(in0, in1, in2, in3, in_off, y0,     sbuf[1], tid);
    wait_async_le2();           // halo row landed
    __syncthreads();
    for (int x = tid; x < OW; x += BLOCK) {
      float A0 = 0.f, B0 = 0.f, C0 = 0.f, D0 = 0.f;
      float A1 = 0.f, B1 = 0.f, C1 = 0.f, D1 = 0.f;
      if (x < W)  { A0 = sbuf[0][0][x];   B0 = sbuf[0][1][x];
                    C0 = sbuf[0][2][x];   D0 = sbuf[0][3][x]; }
      if (x > 0)  { A1 = sbuf[0][0][x-1]; B1 = sbuf[0][1][x-1];
                    C1 = sbuf[0][2][x-1]; D1 = sbuf[0][3][x-1]; }
      float g2 = A0 - 0.5f*B0 + 0.5f*C0 + 0.5f*D0;
      float g3 = A1 + 0.5f*B1 + 0.5f*C1 - 0.5f*D1;
      tprev[x] = g2 + g3;
    }
    __syncthreads();            // buf0 free for reuse
  } else {
    stage_row(in0, in1, in2, in3, in_off, y0, sbuf[1], tid);  // ASYNCcnt = 2/wave
    for (int x = tid; x < OW; x += BLOCK) tprev[x] = 0.f;     // owner-only, no barrier needed
  }

  for (int y = y0; y < y1; ++y) {
    const bool do_stage = (y + 1 < y1) && (y + 1 < H);
    if (do_stage) {
      stage_row(in0, in1, in2, in3, in_off, y + 1, sbuf[cur ^ 1], tid);
      wait_async_le2();         // oldest 2 (row y) complete
    } else {
      wait_async_le0();         // drain
    }
    __syncthreads();            // row y visible to all waves

    const float ry = (y == 0 || y == OW - 1) ? 1.0f : 0.5f;
    float* outrow = out + out_off + (size_t)y * OW;

    if (y < H) {
      const float (*sb)[W] = sbuf[cur];
      for (int x = tid; x < OW; x += BLOCK) {
        float A0 = 0.f, B0 = 0.f, C0 = 0.f, D0 = 0.f;
        float A1 = 0.f, B1 = 0.f, C1 = 0.f, D1 = 0.f;
        if (x < W)  { A0 = sb[0][x];   B0 = sb[1][x];
                      C0 = sb[2][x];   D0 = sb[3][x]; }
        if (x > 0)  { A1 = sb[0][x-1]; B1 = sb[1][x-1];
                      C1 = sb[2][x-1]; D1 = sb[3][x-1]; }
        // GK_inv rows: g_p = A ± .5B ± .5C ± .5D (Hadamard signs)
        float g0 = A0 - 0.5f*B0 - 0.5f*C0 - 0.5f*D0;
        float g1 = A1 + 0.5f*B1 - 0.5f*C1 + 0.5f*D1;
        float g2 = A0 - 0.5f*B0 + 0.5f*C0 + 0.5f*D0;
        float g3 = A1 + 0.5f*B1 + 0.5f*C1 - 0.5f*D1;
        float s  = g0 + g1;                       // s(y,x)
        float rx = (x == 0 || x == OW - 1) ? 1.0f : 0.5f;
        float v  = (s + tprev[x]) * (rx * ry);    // exact: counts in {1,2,4}
        __builtin_nontemporal_store(v, &outrow[x]);
        tprev[x] = g2 + g3;                       // t(y,x) for next row
      }
    } else {  // y == 512: only the t-contribution from row 511 exists
      for (int x = tid; x < OW; x += BLOCK) {
        float rx = (x == 0 || x == OW - 1) ? 1.0f : 0.5f;
        __builtin_nontemporal_store(tprev[x] * (rx * ry), &outrow[x]);
      }
    }
    __syncthreads();            // protect buf `cur` before it is re-staged
    cur ^= 1;
  }
}

extern "C" void kernel_launch(void* const* d_in, const int* in_sizes, int n_in,
                              void* d_out, int out_size, void* d_ws, size_t ws_size,
                              hipStream_t stream) {
  (void)in_sizes; (void)n_in; (void)out_size; (void)d_ws; (void)ws_size;
  const float* in0 = (const float*)d_in[0];
  const float* in1 = (const float*)d_in[1];
  const float* in2 = (const float*)d_in[2];
  const float* in3 = (const float*)d_in[3];
  float* out = (float*)d_out;
  dim3 grid(NPLANES * NBANDS);   // 4096 blocks: 256 (b,c)-planes x 16 row bands
  dim3 block(BLOCK);
  Features2im_kernel<<<grid, block, 0, stream>>>(in0, in1, in2, in3, out);
}